// SPE_2147483648150
// MI455X (gfx1250) — compile-verified
//
#include <hip/hip_runtime.h>

typedef float v2f __attribute__((ext_vector_type(2)));
typedef float v4f __attribute__((ext_vector_type(4)));
typedef float v8f __attribute__((ext_vector_type(8)));

#define BATCH   4
#define CIN     64
#define COUT    64
#define NPTS    40960
#define KNN     16
#define PTOT    (BATCH * NPTS)              // 163840 flattened points
#define ELEMS_PER_CH ((float)(PTOT * (KNN + 1)))
#define NEG_SLOPE 0.2f
#define BN_EPS    1e-6f

// ---------------- ws layout ----------------
// z   : PTOT * 64 floats   (z[p][o], point-major so gathers read 256B runs)
// cnt : PTOT uints         (1 + neighbor occurrences)
// acc : 256 floats         (S1[64], S2[64], s[64], t[64])

__global__ void init_ws(unsigned* __restrict__ cnt, float* __restrict__ acc) {
    int i = blockIdx.x * blockDim.x + threadIdx.x;
    if (i < PTOT) cnt[i] = 1u;        // self-contribution (k==0 column)
    if (i < 256)  acc[i] = 0.0f;
}

__global__ void hist_kernel(const int* __restrict__ nidx, unsigned* __restrict__ cnt) {
    int i = blockIdx.x * blockDim.x + threadIdx.x;      // over B*N*K
    if (i >= PTOT * KNN) return;
    int b = i / (NPTS * KNN);
    int j = nidx[i];                                    // index within batch b
    atomicAdd(&cnt[b * NPTS + j], 1u);
}

// z[p][o] = sum_c W[o][c] * feature[b][c][n],  p = b*NPTS + n
// One wave computes a 16-point x 64-channel strip via V_WMMA_F32_16X16X4_F32.
__global__ void __launch_bounds__(256)
gemm_wmma(const float* __restrict__ feat, const float* __restrict__ W,
          float* __restrict__ z) {
    const int lane   = threadIdx.x & 31;
    const int wave   = blockIdx.x * (blockDim.x >> 5) + (threadIdx.x >> 5);
    const int nwaves = gridDim.x * (blockDim.x >> 5);
    const int half   = lane >> 4;       // K half: lanes 0-15 hold K=0,1; 16-31 hold K=2,3
    const int ml     = lane & 15;       // M (A) / N (B,D) position

    const int ntiles = PTOT / 16;       // 10240; tiles never straddle a batch (40960%16==0)
    for (int tile = wave; tile < ntiles; tile += nwaves) {
        const int p0 = tile * 16;
        const int b  = p0 / NPTS;
        const int n0 = p0 % NPTS;

        // Preload A frags for all 16 K-chunks (reused across 4 o-tiles).
        // A[m][k] = feature[b][c0+k][n0+m]; reg0=K{0|2}, reg1=K{1|3}.
        v2f a[16];
#pragma unroll
        for (int kc = 0; kc < 16; ++kc) {
            const int c = kc * 4 + half * 2;
            const float* fp = feat + ((size_t)(b * CIN + c)) * NPTS + n0 + ml;
            a[kc][0] = fp[0];
            a[kc][1] = fp[NPTS];
        }

#pragma unroll
        for (int ot = 0; ot < 4; ++ot) {
            const int o0 = ot * 16;
            v8f acc = {};
#pragma unroll
            for (int kc = 0; kc < 16; ++kc) {
                const int c = kc * 4 + half * 2;
                // B[k][n] = W[o0+n][c0+k]; row K striped across lanes.
                const float* wp = W + (size_t)(o0 + ml) * CIN + c;
                v2f bf;
                bf[0] = wp[0];
                bf[1] = wp[1];
                acc = __builtin_amdgcn_wmma_f32_16x16x4_f32(
                          false, a[kc], false, bf, (short)0, acc, false, false);
            }
            // D layout: VGPR v, lanes 0-15 -> M=v, lanes 16-31 -> M=v+8; N=ml.
#pragma unroll
            for (int v = 0; v < 8; ++v) {
                const int m = v + half * 8;
                z[(size_t)(p0 + m) * COUT + o0 + ml] = acc[v];
            }
        }
    }
}

// Occurrence-weighted per-channel sums: S1 = sum cnt*z, S2 = sum cnt*z^2
// b128 loads: lane covers 4 channels, 16 points in flight per block-iter.
__global__ void __launch_bounds__(256)
stats_kernel(const float* __restrict__ z, const unsigned* __restrict__ cnt,
             float* __restrict__ S1, float* __restrict__ S2) {
    __shared__ v4f r1[256];
    __shared__ v4f r2[256];
    const int c4 = threadIdx.x & 15;     // channels c4*4 .. c4*4+3
    const int q  = threadIdx.x >> 4;     // 16 points in parallel
    const v4f* zv = (const v4f*)z;
    v4f s1 = {}, s2 = {};
    for (int p = blockIdx.x * 16 + q; p < PTOT; p += gridDim.x * 16) {
        const float w = (float)cnt[p];
        v4f v = zv[(size_t)p * 16 + c4];
#pragma unroll
        for (int j = 0; j < 4; ++j) {
            s1[j] = fmaf(w, v[j], s1[j]);
            s2[j] = fmaf(w * v[j], v[j], s2[j]);
        }
    }
    r1[threadIdx.x] = s1;
    r2[threadIdx.x] = s2;
    __syncthreads();
    if (q == 0) {
#pragma unroll
        for (int qq = 1; qq < 16; ++qq) {
            v4f a = r1[c4 + 16 * qq];
            v4f b = r2[c4 + 16 * qq];
#pragma unroll
            for (int j = 0; j < 4; ++j) { s1[j] += a[j]; s2[j] += b[j]; }
        }
#pragma unroll
        for (int j = 0; j < 4; ++j) {
            atomicAdd(&S1[c4 * 4 + j], s1[j]);
            atomicAdd(&S2[c4 * 4 + j], s2[j]);
        }
    }
}

// Fold mean/var/gamma/beta into per-channel affine (s,t).
__global__ void finalize_kernel(const float* __restrict__ S1, const float* __restrict__ S2,
                                const float* __restrict__ gamma, const float* __restrict__ beta,
                                float* __restrict__ sv, float* __restrict__ tv) {
    const int o = threadIdx.x;
    if (o < COUT) {
        const float inv  = 1.0f / ELEMS_PER_CH;
        const float mean = S1[o] * inv;
        const float var  = S2[o] * inv - mean * mean;
        const float s    = gamma[o] * rsqrtf(var + BN_EPS);
        sv[o] = s;
        tv[o] = fmaf(-mean, s, beta[o]);
    }
}

// out[b][o][n] = lrelu(s*z[p]+t) + sum_k lrelu(s*z[b,idx_k]+t)
// Gathers are b128 (4 channels/lane, 16 points in parallel); writes are b128
// (4 consecutive n per lane) via the LDS transpose tile.
__global__ void __launch_bounds__(256)
fuse_out(const float* __restrict__ z, const int* __restrict__ nidx,
         const float* __restrict__ sv, const float* __restrict__ tv,
         float* __restrict__ out) {
    __shared__ float tile[32][COUT + 1];
    __shared__ int   sidx[32 * KNN];
    __shared__ v4f   ssv[16];
    __shared__ v4f   stv[16];

    const int tid = threadIdx.x;
    const int p0  = blockIdx.x * 32;          // 32 points per block, same batch
    const int b   = p0 / NPTS;
    const int n0  = p0 % NPTS;

    if (tid < 16) {
        ssv[tid] = *(const v4f*)(sv + tid * 4);
        stv[tid] = *(const v4f*)(tv + tid * 4);
    }
#pragma unroll
    for (int i = tid; i < 32 * KNN; i += 256)
        sidx[i] = nidx[(size_t)p0 * KNN + i];
    __syncthreads();

    const int c4 = tid & 15;                  // channels c4*4 .. c4*4+3
    const int pg = tid >> 4;                  // 16 points in parallel
    const v4f s4 = ssv[c4];
    const v4f t4 = stv[c4];
    const v4f* zv = (const v4f*)z;
    const size_t zb4 = (size_t)b * NPTS * 16; // batch base, float4 units

#pragma unroll
    for (int pp = 0; pp < 2; ++pp) {
        const int pl = pp * 16 + pg;          // 0..31
        const int p  = p0 + pl;
        v4f acc;
        {
            v4f v = zv[(size_t)p * 16 + c4];  // self (k==0 column)
#pragma unroll
            for (int j = 0; j < 4; ++j) {
                const float y = fmaf(s4[j], v[j], t4[j]);
                acc[j] = (y >= 0.0f) ? y : y * NEG_SLOPE;
            }
        }
#pragma unroll
        for (int k = 0; k < KNN; ++k) {
            const int jdx = sidx[pl * KNN + k];
            v4f g = zv[zb4 + (size_t)jdx * 16 + c4];   // 256B/point, L2-resident
#pragma unroll
            for (int j = 0; j < 4; ++j) {
                const float y = fmaf(s4[j], g[j], t4[j]);
                acc[j] += (y >= 0.0f) ? y : y * NEG_SLOPE;
            }
        }
#pragma unroll
        for (int j = 0; j < 4; ++j) tile[pl][c4 * 4 + j] = acc[j];
    }
    __syncthreads();

    // transposed b128 writes: lane packs 4 consecutive n for one channel
    const int plq = tid & 7;                  // n-quad 0..7 (32 points)
    const int oh  = tid >> 3;                 // 0..31
#pragma unroll
    for (int oc = 0; oc < 2; ++oc) {
        const int oo = oc * 32 + oh;
        v4f w;
#pragma unroll
        for (int j = 0; j < 4; ++j) w[j] = tile[plq * 4 + j][oo];
        *(v4f*)&out[((size_t)(b * COUT + oo)) * NPTS + n0 + plq * 4] = w;
    }
}

extern "C" void kernel_launch(void* const* d_in, const int* in_sizes, int n_in,
                              void* d_out, int out_size, void* d_ws, size_t ws_size,
                              hipStream_t stream) {
    const float* feat  = (const float*)d_in[0];   // [B,64,N,1]
    const float* W     = (const float*)d_in[1];   // [64,64]
    const float* gamma = (const float*)d_in[2];   // [64]
    const float* beta  = (const float*)d_in[3];   // [64]
    const int*   nidx  = (const int*)d_in[4];     // [B,N,K]
    float*       out   = (float*)d_out;           // [B,64,N,1]

    float*    z   = (float*)d_ws;                         // PTOT*64 floats
    unsigned* cnt = (unsigned*)(z + (size_t)PTOT * COUT); // PTOT uints
    float*    acc = (float*)(cnt + PTOT);                 // 256 floats
    float* S1 = acc;
    float* S2 = acc + 64;
    float* sv = acc + 128;
    float* tv = acc + 192;

    init_ws<<<(PTOT + 255) / 256, 256, 0, stream>>>(cnt, acc);
    hist_kernel<<<(PTOT * KNN + 255) / 256, 256, 0, stream>>>(nidx, cnt);
    gemm_wmma<<<640, 256, 0, stream>>>(feat, W, z);       // 5120 waves, 2 tiles/wave
    stats_kernel<<<1024, 256, 0, stream>>>(z, cnt, S1, S2);
    finalize_kernel<<<1, 64, 0, stream>>>(S1, S2, gamma, beta, sv, tv);
    fuse_out<<<PTOT / 32, 256, 0, stream>>>(z, nidx, sv, tv, out);
}